// CAFI_3367254360368
// MI455X (gfx1250) — compile-verified
//
#include <hip/hip_runtime.h>
#include <math.h>

#define Bq 32
#define Cc 512
#define Ll 1536
#define PATCH 12
#define RANK 4
#define STEPS 127          // L/PATCH - 1
#define EPSf 1e-5f
#define INV_SCALE 0.5f     // 1/sqrt(RANK)
#define NWAVE 4            // waves per attention block

typedef __attribute__((ext_vector_type(2))) float v2f;
typedef __attribute__((ext_vector_type(8))) float v8f;

__device__ __forceinline__ float bn_apply(float x, float g, float b, float m, float v) {
    return (x - m) * (g * rsqrtf(v + EPSf)) + b;
}
__device__ __forceinline__ float gelu_exact(float x) {
    return 0.5f * x * (1.0f + erff(x * 0.7071067811865475f));
}
__device__ __forceinline__ v8f wmma_f32_16x16x4(v2f a, v2f b, v8f c) {
    return __builtin_amdgcn_wmma_f32_16x16x4_f32(false, a, false, b, (short)0, c, false, false);
}

// ---------------------------------------------------------------------------
// Init: first = bn0(x[:, :, :PATCH]) written into d_out slice 0 (scan state).
// ---------------------------------------------------------------------------
__global__ void cafi_init_first(const float* __restrict__ x,
                                const float* __restrict__ bn0g, const float* __restrict__ bn0b,
                                const float* __restrict__ bn0m, const float* __restrict__ bn0v,
                                float* __restrict__ out) {
    int idx = blockIdx.x * blockDim.x + threadIdx.x;   // over B*C*PATCH
    if (idx >= Bq * Cc * PATCH) return;
    int p = idx % PATCH;
    int c = (idx / PATCH) % Cc;
    int b = idx / (PATCH * Cc);
    size_t f0 = (size_t)c * Ll + p;
    size_t xi = ((size_t)b * Cc + c) * Ll + p;
    out[xi] = bn_apply(x[xi], bn0g[f0], bn0b[f0], bn0m[f0], bn0v[f0]);
}

// ---------------------------------------------------------------------------
// Step kernel 1: one block per batch. BN1 -> agg matmul + GELU -> +bn0(x_c)
// -> BN2 -> batch-wide LN reduction -> LN affine -> Q/K/V projections.
// Also prefetches step s+2's input slice into cache (global_prefetch_b8).
// ---------------------------------------------------------------------------
__global__ void __launch_bounds__(256)
cafi_step_prep(const float* __restrict__ x,
               const float* __restrict__ bn0g, const float* __restrict__ bn0b,
               const float* __restrict__ bn0m, const float* __restrict__ bn0v,
               const float* __restrict__ bn1g, const float* __restrict__ bn1b,
               const float* __restrict__ bn1m, const float* __restrict__ bn1v,
               const float* __restrict__ bn2g, const float* __restrict__ bn2b,
               const float* __restrict__ bn2m, const float* __restrict__ bn2v,
               const float* __restrict__ agg_w, const float* __restrict__ agg_b,
               const float* __restrict__ q_w, const float* __restrict__ q_b,
               const float* __restrict__ k_w, const float* __restrict__ k_b,
               const float* __restrict__ v_w, const float* __restrict__ v_b,
               const float* __restrict__ ln_g, const float* __restrict__ ln_b,
               const float* __restrict__ out_state,   // d_out, prev slice at PATCH*s
               float* __restrict__ tmp, float* __restrict__ xt,
               float* __restrict__ Qb, float* __restrict__ Kb, float* __restrict__ Vb,
               int s)
{
    const int b = blockIdx.x;
    const int tid = threadIdx.x;
    __shared__ float rs[256];
    __shared__ float rq[256];

    float lsum = 0.f, lsq = 0.f;
    for (int c = tid; c < Cc; c += 256) {
        float ch[PATCH];
        const float* prev = out_state + ((size_t)b * Cc + c) * Ll + (size_t)PATCH * s;
        #pragma unroll
        for (int p = 0; p < PATCH; ++p) {
            int f = c * PATCH + p;
            ch[p] = bn_apply(prev[p], bn1g[f], bn1b[f], bn1m[f], bn1v[f]);
        }
        size_t tbase = ((size_t)b * Cc + c) * PATCH;
        #pragma unroll
        for (int p = 0; p < PATCH; ++p) {
            float acc = agg_b[p];
            #pragma unroll
            for (int j = 0; j < PATCH; ++j) acc += ch[j] * agg_w[p * PATCH + j];
            float gl = gelu_exact(acc);
            size_t li = (size_t)PATCH * (s + 1) + p;
            size_t f0 = (size_t)c * Ll + li;
            float xv = x[((size_t)b * Cc + c) * Ll + li];
            float xc = bn_apply(xv, bn0g[f0], bn0b[f0], bn0m[f0], bn0v[f0]);
            float tm = gl + xc;
            tmp[tbase + p] = tm;
            int f2 = c * PATCH + p;
            float tv = bn_apply(tm, bn2g[f2], bn2b[f2], bn2m[f2], bn2v[f2]);
            xt[tbase + p] = tv;            // pre-LN value, overwritten below
            lsum += tv;
            lsq  += tv * tv;
        }
        // Hide HBM latency across the serialized scan: pull step s+2's x-slice
        // (and its bn0 params) toward the caches while this step computes.
        if (s + 2 <= STEPS) {
            size_t li2 = (size_t)PATCH * (s + 2);
            __builtin_prefetch(&x[((size_t)b * Cc + c) * Ll + li2], 0, 1);
            __builtin_prefetch(&bn0g[(size_t)c * Ll + li2], 0, 1);
        }
    }
    rs[tid] = lsum; rq[tid] = lsq;
    __syncthreads();
    for (int off = 128; off > 0; off >>= 1) {
        if (tid < off) { rs[tid] += rs[tid + off]; rq[tid] += rq[tid + off]; }
        __syncthreads();
    }
    const float inv_n = 1.0f / (float)(Cc * PATCH);
    const float mu   = rs[0] * inv_n;
    const float var  = rq[0] * inv_n - mu * mu;
    const float rstd = rsqrtf(var + EPSf);

    for (int c = tid; c < Cc; c += 256) {
        float xn[PATCH];
        size_t base = ((size_t)b * Cc + c) * PATCH;
        #pragma unroll
        for (int p = 0; p < PATCH; ++p) {
            float tv = xt[base + p];
            xn[p] = (tv - mu) * rstd * ln_g[p * Cc + c] + ln_b[p * Cc + c];
        }
        #pragma unroll
        for (int p = 0; p < PATCH; ++p) xt[base + p] = xn[p];
        size_t qbase = ((size_t)b * Cc + c) * RANK;
        #pragma unroll
        for (int r = 0; r < RANK; ++r) {
            float q = q_b[r], k = k_b[r];
            #pragma unroll
            for (int j = 0; j < PATCH; ++j) {
                q += xn[j] * q_w[r * PATCH + j];
                k += xn[j] * k_w[r * PATCH + j];
            }
            Qb[qbase + r] = q;
            Kb[qbase + r] = k;
        }
        #pragma unroll
        for (int p = 0; p < PATCH; ++p) {
            float vv = v_b[p];
            #pragma unroll
            for (int j = 0; j < PATCH; ++j) vv += xn[j] * v_w[p * PATCH + j];
            Vb[base + p] = vv;
        }
    }
}

// ---------------------------------------------------------------------------
// Step kernel 2: 4 waves per block; each wave owns a 16-row query tile.
// V (512x12, padded to N=16) staged once per block in LDS and shared by all
// waves. QK^T and P@V via v_wmma_f32_16x16x4_f32 (K=4 == RANK). Per-row
// softmax over 512 cols in a per-wave 16x512 LDS tile. 160 KB LDS/workgroup
// (legal: CDNA5 allows up to 320 KB per workgroup).
// ---------------------------------------------------------------------------
__global__ void __launch_bounds__(NWAVE * 32)
cafi_step_attn(const float* __restrict__ Qg, const float* __restrict__ Kg,
               const float* __restrict__ Vg, const float* __restrict__ xt,
               const float* __restrict__ tmp,
               const float* __restrict__ alpha_p, const float* __restrict__ beta_p,
               float* __restrict__ d_out, int s)
{
    __shared__ float Vl[Cc * 16];            // 32 KB: V padded to 16 cols
    __shared__ float Sm[NWAVE][16 * Cc];     // 128 KB: per-wave score tile

    const int blk  = blockIdx.x;
    const int b    = blk >> 3;               // / (Cc/64)
    const int tid  = threadIdx.x;            // 0..127
    const int wave = tid >> 5;
    const int lane = tid & 31;
    const int c0   = (blk & 7) * 64 + wave * 16;   // query-row tile base
    const int M    = lane & 15;              // A-frag row (also B-frag col)
    const int kh   = (lane >> 4) << 1;       // K offset: 0 (lanes 0-15) / 2 (16-31)
    const int hi   = (lane >> 4) << 3;       // C/D frag row offset: 0 / 8
    float* S = &Sm[wave][0];

    // ---- stage V for this batch into LDS, zero-pad cols 12..15 ----
    {
        const float* Vsrc = Vg + (size_t)b * Cc * PATCH;
        for (int idx = tid; idx < Cc * PATCH; idx += NWAVE * 32) {
            int k = idx / PATCH;
            int n = idx - k * PATCH;
            Vl[k * 16 + n] = Vsrc[idx];
        }
        for (int idx = tid; idx < Cc * 4; idx += NWAVE * 32) {
            int k = idx >> 2;
            Vl[k * 16 + PATCH + (idx & 3)] = 0.f;
        }
    }
    __syncthreads();

    // A fragment: Q rows c0..c0+15, K=4 split across lane halves (ISA 7.12.2)
    v2f afrag;
    {
        const float* Qrow = Qg + ((size_t)b * Cc + c0 + M) * RANK;
        afrag.x = Qrow[kh];
        afrag.y = Qrow[kh + 1];
    }

    // ---- logits: S[16x512] = (Q K^T) * INV_SCALE ----
    for (int j = 0; j < Cc / 16; ++j) {
        int n0 = j * 16;
        v2f bfrag;                           // B[k][n] = K[n0+n][k]
        const float* Krow = Kg + ((size_t)b * Cc + n0 + M) * RANK;
        bfrag.x = Krow[kh];
        bfrag.y = Krow[kh + 1];
        v8f cz = {};
        cz = wmma_f32_16x16x4(afrag, bfrag, cz);
        #pragma unroll
        for (int r = 0; r < 8; ++r)
            S[(r + hi) * Cc + n0 + M] = cz[r] * INV_SCALE;
    }
    __syncthreads();

    // ---- softmax per row: lane handles row M, half (lane>>4) of 512 cols ----
    {
        float* Sr = S + M * Cc + (lane >> 4) * (Cc / 2);
        float mx = -3.402823466e38f;
        for (int i = 0; i < Cc / 2; ++i) mx = fmaxf(mx, Sr[i]);
        mx = fmaxf(mx, __shfl_xor(mx, 16));
        float sum = 0.f;
        for (int i = 0; i < Cc / 2; ++i) {
            float e = __expf(Sr[i] - mx);
            Sr[i] = e;
            sum += e;
        }
        sum += __shfl_xor(sum, 16);
        float inv = 1.0f / sum;
        for (int i = 0; i < Cc / 2; ++i) Sr[i] *= inv;
    }
    __syncthreads();

    // ---- O[16x16] = P[16x512] @ Vpad[512x16], K accumulated 4 at a time ----
    v8f acc = {};
    const int n = lane & 15;                 // output column (12 real + 4 pad)
    for (int k0 = 0; k0 < Cc; k0 += 4) {
        v2f pa, vb;
        pa.x = S[M * Cc + k0 + kh];
        pa.y = S[M * Cc + k0 + kh + 1];
        vb.x = Vl[(k0 + kh) * 16 + n];
        vb.y = Vl[(k0 + kh + 1) * 16 + n];
        acc = wmma_f32_16x16x4(pa, vb, acc);
    }

    // ---- out = tmp + xt + alpha*(A@V) + (1+beta)*V, into d_out slice s+1 ----
    const float alpha = alpha_p[0];
    const float beta1 = 1.0f + beta_p[0];
    if (n < PATCH) {
        #pragma unroll
        for (int r = 0; r < 8; ++r) {
            int c = c0 + r + hi;
            size_t base = ((size_t)b * Cc + c) * PATCH + n;
            float vv = Vl[c * 16 + n];
            float o = tmp[base] + xt[base] + alpha * acc[r] + beta1 * vv;
            d_out[((size_t)b * Cc + c) * Ll + (size_t)PATCH * (s + 1) + n] = o;
        }
    }
}

// ---------------------------------------------------------------------------
extern "C" void kernel_launch(void* const* d_in, const int* in_sizes, int n_in,
                              void* d_out, int out_size, void* d_ws, size_t ws_size,
                              hipStream_t stream) {
    const float* x    = (const float*)d_in[0];
    const float* bn0g = (const float*)d_in[1];
    const float* bn0b = (const float*)d_in[2];
    const float* bn0m = (const float*)d_in[3];
    const float* bn0v = (const float*)d_in[4];
    const float* bn1g = (const float*)d_in[5];
    const float* bn1b = (const float*)d_in[6];
    const float* bn1m = (const float*)d_in[7];
    const float* bn1v = (const float*)d_in[8];
    const float* bn2g = (const float*)d_in[9];
    const float* bn2b = (const float*)d_in[10];
    const float* bn2m = (const float*)d_in[11];
    const float* bn2v = (const float*)d_in[12];
    const float* agg_w = (const float*)d_in[13];
    const float* agg_b = (const float*)d_in[14];
    const float* q_w   = (const float*)d_in[15];
    const float* q_b   = (const float*)d_in[16];
    const float* k_w   = (const float*)d_in[17];
    const float* k_b   = (const float*)d_in[18];
    const float* v_w   = (const float*)d_in[19];
    const float* v_b   = (const float*)d_in[20];
    const float* ln_g  = (const float*)d_in[21];
    const float* ln_b  = (const float*)d_in[22];
    const float* alpha = (const float*)d_in[23];
    const float* beta  = (const float*)d_in[24];

    float* out = (float*)d_out;
    float* ws  = (float*)d_ws;
    float* tmp = ws;                                  // B*C*12
    float* xt  = tmp + (size_t)Bq * Cc * PATCH;       // B*C*12
    float* Qb  = xt  + (size_t)Bq * Cc * PATCH;       // B*C*4
    float* Kb  = Qb  + (size_t)Bq * Cc * RANK;        // B*C*4
    float* Vb  = Kb  + (size_t)Bq * Cc * RANK;        // B*C*12

    const int n0 = Bq * Cc * PATCH;
    cafi_init_first<<<(n0 + 255) / 256, 256, 0, stream>>>(x, bn0g, bn0b, bn0m, bn0v, out);

    for (int s = 0; s < STEPS; ++s) {
        cafi_step_prep<<<Bq, 256, 0, stream>>>(
            x, bn0g, bn0b, bn0m, bn0v, bn1g, bn1b, bn1m, bn1v,
            bn2g, bn2b, bn2m, bn2v, agg_w, agg_b,
            q_w, q_b, k_w, k_b, v_w, v_b, ln_g, ln_b,
            out, tmp, xt, Qb, Kb, Vb, s);
        cafi_step_attn<<<Bq * (Cc / 64), NWAVE * 32, 0, stream>>>(
            Qb, Kb, Vb, xt, tmp, alpha, beta, out, s);
    }
}